// GraphLogLikelihood_3865470566400
// MI455X (gfx1250) — compile-verified
//
#include <hip/hip_runtime.h>
#include <math.h>

// GraphLogLikelihood on gfx1250:
//   e_dot[e]  = <input[src_e], input[dst_e]>  (64-dim f32)
//   out = sum_edges log1p(-exp(-e_dot)) - sum_nonedges ne_dot
//
// 16 pairs per wave-tile; diagonal of A(16x64) x B(64x16) via 16x
// V_WMMA_F32_16X16X4_F32 (full f32 precision). Problem is L2-gather bound
// (12.8MB matrix is L2-resident), so gathers use b128 (one 16B K-chunk per
// lane) to halve L2 request count, with a half-wave shfl_xor exchange to
// land data in the WMMA-mandated lane layout (lane L: K%4 in {0,1};
// lane L+16: K%4 in {2,3}).

typedef float v2f __attribute__((ext_vector_type(2)));
typedef float v4f __attribute__((ext_vector_type(4)));
typedef float v8f __attribute__((ext_vector_type(8)));

#define NCOMM 64

// MODE 0: edges, f(x) = log(-expm1(-x));  MODE 1: non-edges, f(x) = -x
template <int MODE>
__global__ __launch_bounds__(256) void pair_dot_kernel(
    const float* __restrict__ inp,
    const int*   __restrict__ src,
    const int*   __restrict__ dst,
    int npairs,
    float* __restrict__ blockPartials)
{
    __shared__ float s_wave[8];

    const int lane = threadIdx.x & 31;
    const int wave = threadIdx.x >> 5;
    const int lr   = lane & 15;   // pair slot within tile this lane loads for
    const int half = lane >> 4;   // half-wave id (selects K sub-chunk)

    const int wavesInGrid = gridDim.x * 8;
    const int ntiles = (npairs + 15) >> 4;

    float acc = 0.0f;

    for (int t = blockIdx.x * 8 + wave; t < ntiles; t += wavesInGrid) {
        int e  = t * 16 + lr;
        int ce = (e < npairs) ? e : 0;            // clamp tail (masked later)
        int sr = src[ce];
        int dr = dst[ce];

        // Lane loads 16B chunks at K base = 8*kk + 4*half (16B aligned).
        const float* ap = inp + sr * NCOMM + 4 * half;
        const float* bp = inp + dr * NCOMM + 4 * half;

        v8f c = {0.f, 0.f, 0.f, 0.f, 0.f, 0.f, 0.f, 0.f};
#pragma unroll
        for (int kk = 0; kk < 8; ++kk) {
            v4f a = *(const v4f*)(ap + 8 * kk);    // global_load_b128
            v4f b = *(const v4f*)(bp + 8 * kk);    // global_load_b128

            // Half-wave exchange: each lane sends the pair its partner
            // (lane^16) needs. half-0 sends .zw (partner's step-s0 data),
            // half-1 sends .xy (partner's step-s1 data).
            float as0 = half ? a.x : a.z;
            float as1 = half ? a.y : a.w;
            float bs0 = half ? b.x : b.z;
            float bs1 = half ? b.y : b.w;
            float ar0 = __shfl_xor(as0, 16, 32);
            float ar1 = __shfl_xor(as1, 16, 32);
            float br0 = __shfl_xor(bs0, 16, 32);
            float br1 = __shfl_xor(bs1, 16, 32);

            // Step s0: K = 8kk..8kk+3. half-0 supplies own .xy (K%4 in {0,1}),
            // half-1 supplies partner's .zw (K%4 in {2,3}).
            v2f a0; a0.x = half ? ar0 : a.x;  a0.y = half ? ar1 : a.y;
            v2f b0; b0.x = half ? br0 : b.x;  b0.y = half ? br1 : b.y;
            // Step s1: K = 8kk+4..8kk+7. half-0 supplies partner's .xy,
            // half-1 supplies own .zw.
            v2f a1; a1.x = half ? a.z : ar0;  a1.y = half ? a.w : ar1;
            v2f b1; b1.x = half ? b.z : br0;  b1.y = half ? b.w : br1;

            c = __builtin_amdgcn_wmma_f32_16x16x4_f32(
                    false, a0, false, b0, (short)0, c, false, false);
            c = __builtin_amdgcn_wmma_f32_16x16x4_f32(
                    false, a1, false, b1, (short)0, c, false, false);
        }

        // Diagonal of C (16x16 f32): lanes 0-15 hold (M=r, N=lane), lanes
        // 16-31 hold (M=8+r, N=lane-16). Diag e<8 -> lane e, vgpr e;
        // diag e>=8 -> lane e+16, vgpr e-8. Branch-free extraction.
        bool isdiag = (lane < 8) || (lane >= 24);
        int  ridx   = (lane < 8) ? lane : (lane - 24);
        float d = 0.0f;
#pragma unroll
        for (int r = 0; r < 8; ++r)
            d = (isdiag && (ridx == r)) ? c[r] : d;

        int ev = t * 16 + ((lane < 8) ? lane : (lane - 16));
        if (isdiag && ev < npairs) {
            if (MODE == 0) {
                // log(1 - exp(-x)) computed accurately as log(-expm1(-x))
                acc += logf(-expm1f(-d));
            } else {
                acc -= d;   // non-edge contribution enters with minus sign
            }
        }
    }

    // Wave32 reduction (fixed order -> deterministic)
#pragma unroll
    for (int off = 16; off > 0; off >>= 1)
        acc += __shfl_xor(acc, off, 32);

    if (lane == 0) s_wave[wave] = acc;
    __syncthreads();

    if (threadIdx.x == 0) {
        float s = 0.0f;
#pragma unroll
        for (int w = 0; w < 8; ++w) s += s_wave[w];
        blockPartials[blockIdx.x] = s;
    }
}

__global__ __launch_bounds__(256) void final_reduce_kernel(
    const float* __restrict__ ws, int n, float* __restrict__ out)
{
    __shared__ float s[256];
    float a = 0.0f;
    for (int i = threadIdx.x; i < n; i += 256) a += ws[i];
    s[threadIdx.x] = a;
    __syncthreads();
#pragma unroll
    for (int off = 128; off > 0; off >>= 1) {
        if (threadIdx.x < off) s[threadIdx.x] += s[threadIdx.x + off];
        __syncthreads();
    }
    if (threadIdx.x == 0) out[0] = s[0];
}

extern "C" void kernel_launch(void* const* d_in, const int* in_sizes, int n_in,
                              void* d_out, int out_size, void* d_ws, size_t ws_size,
                              hipStream_t stream)
{
    const float* inp = (const float*)d_in[0];
    const int*   ei  = (const int*)d_in[1];   // [2, E] flat: row0=src, row1=dst
    const int*   nei = (const int*)d_in[2];   // [2, NE]
    const int E  = in_sizes[1] / 2;
    const int NE = in_sizes[2] / 2;

    float* ws = (float*)d_ws;
    const int BE = 512;    // edge blocks     (62,500 tiles over 4096 waves)
    const int BN = 1024;   // non-edge blocks (250,000 tiles over 8192 waves)

    pair_dot_kernel<0><<<BE, 256, 0, stream>>>(inp, ei,      ei  + E,  E,  ws);
    pair_dot_kernel<1><<<BN, 256, 0, stream>>>(inp, nei,     nei + NE, NE, ws + BE);
    final_reduce_kernel<<<1, 256, 0, stream>>>(ws, BE + BN, (float*)d_out);
}